// SelfAttention_68195490726090
// MI455X (gfx1250) — compile-verified
//
#include <hip/hip_runtime.h>
#include <math.h>

// ---------------------------------------------------------------------------
// Causal multi-head self-attention forward (fp32) for MI455X / gfx1250.
// All GEMMs on V_WMMA_F32_16X16X4_F32 (exact fp32 matrix path, wave32).
// Large GEMMs (QKV, proj) use double-buffered LDS tiles filled with
// global_load_async_to_lds_b128 and s_wait_asynccnt (CDNA5 async path).
//   b=2, n=2048, hidden=1024, heads=16, d=64
// Outputs (concatenated): out [2,2048,1024], attn [2,16,2048,2048]
// ---------------------------------------------------------------------------

typedef __attribute__((ext_vector_type(2))) float v2f;
typedef __attribute__((ext_vector_type(4))) float v4f;
typedef __attribute__((ext_vector_type(8))) float v8f;
typedef __attribute__((ext_vector_type(4))) int   v4i;

#define AS1 __attribute__((address_space(1)))
#define AS3 __attribute__((address_space(3)))

#define B_SZ 2
#define SEQ  2048
#define HID  1024
#define NH   16
#define HD   64
#define MASK_VALUE (-3.402823466e38f)

// Tiled-GEMM geometry (QKV / proj): block = 256 threads = 8 waves,
// tile = 32(M) x 128(N), K staged in chunks of 32 through LDS.
#define BM 32
#define BN 128
#define BK 32
#define BKP 36    // padded A row stride (floats): 16 lanes stride-36 -> 16 banks
#define BNP 136   // padded B row stride (floats): half-wave offset 272 = 16 mod 64

// ---- 16B global -> LDS copy; async on CDNA5, sync fallback otherwise -------
__device__ __forceinline__ void cp16_g2s(float* lds, const float* g) {
#if __has_builtin(__builtin_amdgcn_global_load_async_to_lds_b128)
  __builtin_amdgcn_global_load_async_to_lds_b128(
      (AS1 v4i*)(void*)g, (AS3 v4i*)lds, 0, 0);
#else
  *(v4f*)lds = *(const v4f*)g;
#endif
}

template <int N>
__device__ __forceinline__ void wait_async() {
#if __has_builtin(__builtin_amdgcn_s_wait_asynccnt)
  __builtin_amdgcn_s_wait_asynccnt(N);
#else
  __builtin_amdgcn_fence(__ATOMIC_RELEASE, "workgroup");
#endif
}

// ---------------------------------------------------------------------------
// Block-level 32x128 fp32 WMMA GEMM with double-buffered LDS staging.
// Ag: row-major [BM x K] slice (lda), Bg: row-major [K x BN] slice (ldb).
// Wave w owns N-columns [w*16, w*16+16); computes 2 M-subtiles reusing B.
// ---------------------------------------------------------------------------
__device__ __forceinline__ void gemm_block_32x128(
    const float* __restrict__ Ag, int lda,
    const float* __restrict__ Bg, int ldb, int K,
    float (*As)[BM * BKP], float (*Bs)[BK * BNP], v8f acc[2]) {
  const int t    = threadIdx.x;
  const int lane = t & 31, wave = t >> 5;
  const int l16  = lane & 15, hf = lane >> 4;
  const int cb   = wave * 16 + l16;        // column within the 128-wide tile

  const int arow = t >> 3, aseg = t & 7;   // A chunk: 32 rows x 8 float4

  auto stage = [&](int kc, int buf) {      // 5 async b128 ops per thread
    cp16_g2s(&As[buf][arow * BKP + aseg * 4],
             Ag + (size_t)arow * lda + kc + aseg * 4);
#pragma unroll
    for (int j = 0; j < 4; ++j) {          // B chunk: 32 rows x 32 float4
      const int idx = t + j * 256, row = idx >> 5, seg = idx & 31;
      cp16_g2s(&Bs[buf][row * BNP + seg * 4],
               Bg + (size_t)(kc + row) * ldb + seg * 4);
    }
  };

  stage(0, 0);
  const int NC = K / BK;
  for (int c = 0; c < NC; ++c) {
    const int buf = c & 1;
    if (c + 1 < NC) { stage((c + 1) * BK, buf ^ 1); wait_async<5>(); }
    else            { wait_async<0>(); }
    __syncthreads();
    const float* Asb = As[buf];
    const float* Bsb = Bs[buf];
#pragma unroll
    for (int k = 0; k < BK; k += 4) {
      const int kk = k + 2 * hf;
      v2f b, a0, a1;
      b[0]  = Bsb[kk * BNP + cb];
      b[1]  = Bsb[(kk + 1) * BNP + cb];
      a0[0] = Asb[l16 * BKP + kk];
      a0[1] = Asb[l16 * BKP + kk + 1];
      a1[0] = Asb[(16 + l16) * BKP + kk];
      a1[1] = Asb[(16 + l16) * BKP + kk + 1];
      acc[0] = __builtin_amdgcn_wmma_f32_16x16x4_f32(false, a0, false, b,
                                                     (short)0, acc[0], false, false);
      acc[1] = __builtin_amdgcn_wmma_f32_16x16x4_f32(false, a1, false, b,
                                                     (short)0, acc[1], false, false);
    }
    __syncthreads();
  }
}

// -------- wave-level 16x16 fp32 WMMA tile, A row-major [M,K], B row-major [K,N]
__device__ __forceinline__ v8f wmma_tile_nn(const float* __restrict__ A, int lda,
                                            const float* __restrict__ Bm, int ldb,
                                            int K, int l16, int hf) {
  v8f acc = {};
#pragma unroll 4
  for (int k = 0; k < K; k += 4) {
    const int kk = k + 2 * hf;
    v2f a, b;
    a[0] = A[l16 * lda + kk];
    a[1] = A[l16 * lda + kk + 1];
    b[0] = Bm[(size_t)kk * ldb + l16];
    b[1] = Bm[(size_t)(kk + 1) * ldb + l16];
    acc = __builtin_amdgcn_wmma_f32_16x16x4_f32(false, a, false, b,
                                                (short)0, acc, false, false);
  }
  return acc;
}

// -------- same, but B stored as [N,K] (computes A * B^T)
__device__ __forceinline__ v8f wmma_tile_nt(const float* __restrict__ A, int lda,
                                            const float* __restrict__ Bt, int ldb,
                                            int K, int l16, int hf) {
  v8f acc = {};
#pragma unroll 4
  for (int k = 0; k < K; k += 4) {
    const int kk = k + 2 * hf;
    v2f a, b;
    a[0] = A[l16 * lda + kk];
    a[1] = A[l16 * lda + kk + 1];
    b[0] = Bt[l16 * ldb + kk];
    b[1] = Bt[l16 * ldb + kk + 1];
    acc = __builtin_amdgcn_wmma_f32_16x16x4_f32(false, a, false, b,
                                                (short)0, acc, false, false);
  }
  return acc;
}

// ---------------------------------------------------------------------------
// Kernel 1: fused QKV projection. x[4096,1024] @ w_qkv[1024,3072],
// scattered to split-head q/k/v [b,h,n,d]; q scaled by d^-0.5 = 1/8.
// grid (24, 128), block 256.
// ---------------------------------------------------------------------------
__global__ void k_qkv(const float* __restrict__ x, const float* __restrict__ w,
                      float* __restrict__ q, float* __restrict__ kv,
                      float* __restrict__ vv) {
  __shared__ __align__(16) float As[2][BM * BKP];
  __shared__ __align__(16) float Bs[2][BK * BNP];
  const int nb = blockIdx.x, mb = blockIdx.y;

  v8f acc[2] = {v8f{}, v8f{}};
  gemm_block_32x128(x + (size_t)mb * BM * HID, HID,
                    w + (size_t)nb * BN, 3 * HID, HID, As, Bs, acc);

  const int lane = threadIdx.x & 31, wave = threadIdx.x >> 5;
  const int l16 = lane & 15, hf = lane >> 4;
  const int col   = nb * BN + wave * 16 + l16;
  const int which = col / HID;               // 0=q 1=k 2=v
  const int hidx  = col % HID;
  const int head  = hidx / HD, dd = hidx % HD;
  float* dst = (which == 0) ? q : (which == 1) ? kv : vv;
  const float scale = (which == 0) ? 0.125f : 1.0f;

#pragma unroll
  for (int mi = 0; mi < 2; ++mi) {
#pragma unroll
    for (int i = 0; i < 8; ++i) {
      const int row = mb * BM + mi * 16 + hf * 8 + i;   // [0,4096)
      const int bb = row >> 11, nn = row & (SEQ - 1);
      dst[(((size_t)bb * NH + head) * SEQ + nn) * HD + dd] = acc[mi][i] * scale;
    }
  }
}

// ---------------------------------------------------------------------------
// Kernel 2: scores S = q @ k^T per (b,h) with causal mask, written to attn.
// grid (16, 128, 32), block 256. Fully-masked tiles skip the GEMM entirely.
// ---------------------------------------------------------------------------
__global__ void k_scores(const float* __restrict__ q, const float* __restrict__ kv,
                         float* __restrict__ attn) {
  const int lane = threadIdx.x & 31, wave = threadIdx.x >> 5;
  const int l16 = lane & 15, hf = lane >> 4;
  const int ntile = blockIdx.x * 8 + wave;   // 0..127 (cols)
  const int mtile = blockIdx.y;              // 0..127 (rows)
  const int bh    = blockIdx.z;              // 0..31

  float* S = attn + (size_t)bh * SEQ * SEQ;
  const int col = ntile * 16 + l16;

  if (ntile > mtile) {                       // tile entirely above diagonal
#pragma unroll
    for (int i = 0; i < 8; ++i) {
      const int row = mtile * 16 + hf * 8 + i;
      S[(size_t)row * SEQ + col] = MASK_VALUE;
    }
    return;
  }

  const float* Aq = q  + (size_t)bh * SEQ * HD + (size_t)mtile * 16 * HD;
  const float* Bk = kv + (size_t)bh * SEQ * HD + (size_t)ntile * 16 * HD;
  v8f acc = wmma_tile_nt(Aq, HD, Bk, HD, HD, l16, hf);

#pragma unroll
  for (int i = 0; i < 8; ++i) {
    const int row = mtile * 16 + hf * 8 + i;
    float val = acc[i];
    if (col > row) val = MASK_VALUE;         // only diagonal tiles hit this
    S[(size_t)row * SEQ + col] = val;
  }
}

// ---------------------------------------------------------------------------
// Kernel 3: row softmax over attn, in place. One block per row (65536 rows).
// Masked entries (-FLT_MAX) underflow to exactly 0 in expf, matching reference.
// ---------------------------------------------------------------------------
__global__ void k_softmax(float* __restrict__ attn) {
  const size_t row = blockIdx.x;
  float* p = attn + row * SEQ;
  const int t = threadIdx.x;
  __shared__ float red[256];

  float vals[8];
  float m = MASK_VALUE;
#pragma unroll
  for (int j = 0; j < 8; ++j) {
    vals[j] = p[t + j * 256];
    m = fmaxf(m, vals[j]);
  }
  red[t] = m;
  __syncthreads();
  for (int s = 128; s > 0; s >>= 1) {
    if (t < s) red[t] = fmaxf(red[t], red[t + s]);
    __syncthreads();
  }
  m = red[0];
  __syncthreads();

  float sum = 0.f;
#pragma unroll
  for (int j = 0; j < 8; ++j) {
    vals[j] = __expf(vals[j] - m);
    sum += vals[j];
  }
  red[t] = sum;
  __syncthreads();
  for (int s = 128; s > 0; s >>= 1) {
    if (t < s) red[t] += red[t + s];
    __syncthreads();
  }
  const float inv = 1.0f / red[0];
#pragma unroll
  for (int j = 0; j < 8; ++j) p[t + j * 256] = vals[j] * inv;
}

// ---------------------------------------------------------------------------
// Kernel 4: O = attn @ v per (b,h), written directly into merged [b,n,hidden].
// Causal K-limit: columns beyond the row block are exactly zero after softmax.
// grid (1, 128, 32), block 128 (4 waves -> 4 d-tiles of 16).
// ---------------------------------------------------------------------------
__global__ void k_pv(const float* __restrict__ attn, const float* __restrict__ vv,
                     float* __restrict__ xa) {
  const int lane = threadIdx.x & 31, wave = threadIdx.x >> 5;
  const int l16 = lane & 15, hf = lane >> 4;
  const int ntile = wave;                    // 0..3
  const int mtile = blockIdx.y;              // 0..127
  const int bh    = blockIdx.z;              // 0..31

  const float* Ap = attn + (size_t)bh * SEQ * SEQ + (size_t)mtile * 16 * SEQ;
  const float* Bp = vv   + (size_t)bh * SEQ * HD + ntile * 16;
  const int Klim = (mtile + 1) * 16;         // causality: rest of row is 0

  v8f acc = wmma_tile_nn(Ap, SEQ, Bp, HD, Klim, l16, hf);

  const int b_ = bh >> 4, h_ = bh & 15;
#pragma unroll
  for (int i = 0; i < 8; ++i) {
    const int n = mtile * 16 + hf * 8 + i;
    xa[((size_t)b_ * SEQ + n) * HID + h_ * HD + ntile * 16 + l16] = acc[i];
  }
}

// ---------------------------------------------------------------------------
// Kernel 5: out = xa @ w_proj + b_proj.  grid (8, 128), block 256.
// ---------------------------------------------------------------------------
__global__ void k_proj(const float* __restrict__ xa, const float* __restrict__ w,
                       const float* __restrict__ bias, float* __restrict__ out) {
  __shared__ __align__(16) float As[2][BM * BKP];
  __shared__ __align__(16) float Bs[2][BK * BNP];
  const int nb = blockIdx.x, mb = blockIdx.y;

  v8f acc[2] = {v8f{}, v8f{}};
  gemm_block_32x128(xa + (size_t)mb * BM * HID, HID,
                    w + (size_t)nb * BN, HID, HID, As, Bs, acc);

  const int lane = threadIdx.x & 31, wave = threadIdx.x >> 5;
  const int l16 = lane & 15, hf = lane >> 4;
  const int col = nb * BN + wave * 16 + l16;
  const float bb = bias[col];

#pragma unroll
  for (int mi = 0; mi < 2; ++mi) {
#pragma unroll
    for (int i = 0; i < 8; ++i) {
      const size_t row = (size_t)mb * BM + mi * 16 + hf * 8 + i;
      out[row * HID + col] = acc[mi][i] + bb;
    }
  }
}

// ---------------------------------------------------------------------------
extern "C" void kernel_launch(void* const* d_in, const int* in_sizes, int n_in,
                              void* d_out, int out_size, void* d_ws, size_t ws_size,
                              hipStream_t stream) {
  (void)in_sizes; (void)n_in; (void)out_size; (void)ws_size;
  const float* x      = (const float*)d_in[0];
  const float* w_qkv  = (const float*)d_in[1];
  const float* w_proj = (const float*)d_in[2];
  const float* b_proj = (const float*)d_in[3];

  float* out  = (float*)d_out;                                   // [2,2048,1024]
  float* attn = out + (size_t)B_SZ * SEQ * HID;                  // [2,16,2048,2048]

  const size_t HSZ = (size_t)B_SZ * NH * SEQ * HD;               // 4,194,304
  float* ws = (float*)d_ws;
  float* q  = ws;
  float* kv = ws + HSZ;
  float* vv = ws + 2 * HSZ;
  float* xa = ws + 3 * HSZ;                                      // merged [b,n,hid]

  k_qkv    <<<dim3(24, 128),      256, 0, stream>>>(x, w_qkv, q, kv, vv);
  k_scores <<<dim3(16, 128, 32),  256, 0, stream>>>(q, kv, attn);
  k_softmax<<<dim3(65536),        256, 0, stream>>>(attn);
  k_pv     <<<dim3(1, 128, 32),   128, 0, stream>>>(attn, vv, xa);
  k_proj   <<<dim3(8, 128),       256, 0, stream>>>(xa, w_proj, b_proj, out);
}